// GATNet_38079180046797
// MI455X (gfx1250) — compile-verified
//
#include <hip/hip_runtime.h>
#include <hip/hip_bf16.h>

typedef __attribute__((ext_vector_type(2))) float v2f;
typedef __attribute__((ext_vector_type(8))) float v8f;

#define CDIM 128
#define NEG_SLOPE 0.2f

// ---------------------------------------------------------------------------
// Fused projection GEMM: out = x @ W  (+ bias for the residual projection).
// blockIdx.y selects which of the three weight matrices / outputs.
// 256 threads = 8 waves; each wave computes a 16-row x 128-col output tile
// using V_WMMA_F32_16X16X4_F32 (32 k-steps of K=4 for the C=128 reduction).
// W (128x128 fp32 = 64KB) is staged in LDS once per block.
// ---------------------------------------------------------------------------
__global__ __launch_bounds__(256) void gat_gemm3(
    const float* __restrict__ x,
    const float* __restrict__ Wl, const float* __restrict__ Wr,
    const float* __restrict__ Wres, const float* __restrict__ bias,
    float* __restrict__ outl, float* __restrict__ outr, float* __restrict__ outres,
    int nrows)
{
    __shared__ float Ws[CDIM * CDIM];   // 64KB

    const int z = blockIdx.y;           // 0=Wl, 1=Wr, 2=Wres(+bias)
    const float* W   = (z == 0) ? Wl   : (z == 1) ? Wr   : Wres;
    float*       out = (z == 0) ? outl : (z == 1) ? outr : outres;

    // cooperative LDS stage of W (coalesced float4)
    {
        const float4* Wv = (const float4*)W;
        float4* Sv = (float4*)Ws;
        for (int i = threadIdx.x; i < (CDIM * CDIM) / 4; i += 256) Sv[i] = Wv[i];
    }
    __syncthreads();

    const int wave = threadIdx.x >> 5;
    const int lane = threadIdx.x & 31;
    const int tile = blockIdx.x * 8 + wave;           // 16-row tile index
    if (tile * 16 >= nrows) return;

    const int m     = lane & 15;                      // row within tile / col within tile
    const int khalf = (lane >> 4) * 2;                // 0 for lanes 0-15, 2 for 16-31
    const float* xrow = x + (size_t)(tile * 16 + m) * CDIM;

    // hint the next row tile toward the caches (global_prefetch_b8)
    __builtin_prefetch(xrow + 16 * CDIM, 0, 1);

    // Cache this wave's A fragments: per k-step the lane holds x[m, 4k+khalf .. +1]
    float2 areg[32];
#pragma unroll
    for (int ks = 0; ks < 32; ++ks)
        areg[ks] = *(const float2*)(xrow + 4 * ks + khalf);

    for (int nt = 0; nt < 8; ++nt) {
        const int nbase = nt * 16 + m;
        v8f acc = {};
#pragma unroll
        for (int ks = 0; ks < 32; ++ks) {
            v2f a; a.x = areg[ks].x; a.y = areg[ks].y;
            v2f b;
            b.x = Ws[(4 * ks + khalf)     * CDIM + nbase];
            b.y = Ws[(4 * ks + khalf + 1) * CDIM + nbase];
            acc = __builtin_amdgcn_wmma_f32_16x16x4_f32(
                false, a, false, b, (short)0, acc, false, false);
        }
        // C/D layout: VGPR r -> M = r + 8*(lane>>4), N = lane&15
        const float badd = (z == 2) ? bias[nbase] : 0.0f;
        const int mhi = (lane >> 4) * 8;
        float* orow = out + (size_t)(tile * 16 + mhi) * CDIM + nbase;
#pragma unroll
        for (int r = 0; r < 8; ++r)
            orow[(size_t)r * CDIM] = acc[r] + badd;
    }
}

// ------------------- segment-softmax helpers (ordered-uint float max) -------
__device__ __forceinline__ unsigned f32_ord(float f) {
    unsigned u = __float_as_uint(f);
    return (u & 0x80000000u) ? ~u : (u | 0x80000000u);
}
__device__ __forceinline__ float ord_f32(unsigned u) {
    return __uint_as_float((u & 0x80000000u) ? (u ^ 0x80000000u) : ~u);
}

__global__ __launch_bounds__(256) void gat_init_nodes(unsigned* __restrict__ emax, int n) {
    int i = blockIdx.x * 256 + threadIdx.x;
    if (i < n) emax[i] = 0x007FFFFFu;   // f32_ord(-inf)
}

// One wave per edge: e = leaky_relu(xl[src]+xr[dst]) . att ; segment max into emax[dst]
__global__ __launch_bounds__(256) void gat_edge_logits(
    const float* __restrict__ xl, const float* __restrict__ xr,
    const float* __restrict__ att,
    const int* __restrict__ src, const int* __restrict__ dst,
    float* __restrict__ e, unsigned* __restrict__ emax, int E)
{
    const int edge = blockIdx.x * 8 + (threadIdx.x >> 5);
    if (edge >= E) return;
    const int lane = threadIdx.x & 31;
    const int j = src[edge], i = dst[edge];

    const float4 a = *(const float4*)(xl + (size_t)j * CDIM + lane * 4);
    const float4 b = *(const float4*)(xr + (size_t)i * CDIM + lane * 4);
    const float4 w = *(const float4*)(att + lane * 4);

    float s = 0.f, h;
    h = a.x + b.x; h = (h > 0.f) ? h : NEG_SLOPE * h; s += h * w.x;
    h = a.y + b.y; h = (h > 0.f) ? h : NEG_SLOPE * h; s += h * w.y;
    h = a.z + b.z; h = (h > 0.f) ? h : NEG_SLOPE * h; s += h * w.z;
    h = a.w + b.w; h = (h > 0.f) ? h : NEG_SLOPE * h; s += h * w.w;

#pragma unroll
    for (int off = 16; off > 0; off >>= 1) s += __shfl_xor(s, off, 32);

    if (lane == 0) {
        e[edge] = s;
        atomicMax(&emax[i], f32_ord(s));
    }
}

// Thread per edge: ex = exp(e - emax[dst]); denom[dst] += ex  (e overwritten with ex)
__global__ __launch_bounds__(256) void gat_edge_exp(
    float* __restrict__ e, const unsigned* __restrict__ emax,
    const int* __restrict__ dst, float* __restrict__ denom, int E)
{
    int edge = blockIdx.x * 256 + threadIdx.x;
    if (edge >= E) return;
    const int i = dst[edge];
    const float ex = __expf(e[edge] - ord_f32(emax[i]));
    e[edge] = ex;
    atomicAdd(&denom[i], ex);
}

// One wave per edge: agg[dst] += (ex/denom[dst]) * xl[src]
__global__ __launch_bounds__(256) void gat_edge_scatter(
    const float* __restrict__ e, const float* __restrict__ denom,
    const float* __restrict__ xl,
    const int* __restrict__ src, const int* __restrict__ dst,
    float* __restrict__ agg, int E)
{
    const int edge = blockIdx.x * 8 + (threadIdx.x >> 5);
    if (edge >= E) return;
    const int lane = threadIdx.x & 31;
    const int j = src[edge], i = dst[edge];
    const float alpha = e[edge] / fmaxf(denom[i], 1e-16f);

    const float4 a = *(const float4*)(xl + (size_t)j * CDIM + lane * 4);
    float* p = agg + (size_t)i * CDIM + lane * 4;
    atomicAdd(p + 0, alpha * a.x);
    atomicAdd(p + 1, alpha * a.y);
    atomicAdd(p + 2, alpha * a.z);
    atomicAdd(p + 3, alpha * a.w);
}

// out = relu(agg + xres)   (xres already includes bias)
__global__ __launch_bounds__(256) void gat_finalize(
    const float* __restrict__ agg, const float* __restrict__ xres,
    float* __restrict__ out, long n)
{
    long idx = (long)blockIdx.x * 256 + threadIdx.x;
    if (idx < n) out[idx] = fmaxf(agg[idx] + xres[idx], 0.f);
}

// ---------------------------------------------------------------------------
extern "C" void kernel_launch(void* const* d_in, const int* in_sizes, int n_in,
                              void* d_out, int out_size, void* d_ws, size_t ws_size,
                              hipStream_t stream) {
    const float* x   = (const float*)d_in[0];
    const int*   ei  = (const int*)d_in[1];
    const int N = in_sizes[0] / CDIM;
    const int E = in_sizes[1] / 2;
    const int* src = ei;
    const int* dst = ei + E;

    const float* Wm[2][3]  = { { (const float*)d_in[2], (const float*)d_in[3], (const float*)d_in[5] },
                               { (const float*)d_in[7], (const float*)d_in[8], (const float*)d_in[10] } };
    const float* attv[2]   = { (const float*)d_in[4], (const float*)d_in[9] };
    const float* biasv[2]  = { (const float*)d_in[6], (const float*)d_in[11] };

    const size_t NC = (size_t)N * CDIM;
    float*    xl    = (float*)d_ws;
    float*    xr    = xl   + NC;
    float*    xres  = xr   + NC;
    float*    agg   = xres + NC;
    float*    x1    = agg  + NC;          // layer-1 output
    float*    ebuf  = x1   + NC;          // [E] logits, then exp values
    float*    denom = ebuf + E;           // [N]
    unsigned* emax  = (unsigned*)(denom + N);  // [N]

    const int tiles = (N + 15) / 16;
    const dim3 gGemm((tiles + 7) / 8, 3);
    const int gEdgeWave = (E + 7) / 8;
    const int gEdgeThr  = (E + 255) / 256;
    const int gNode     = (N + 255) / 256;
    const int gElem     = (int)((NC + 255) / 256);

    for (int layer = 0; layer < 2; ++layer) {
        const float* xin = (layer == 0) ? x : x1;
        float* xout = (layer == 0) ? x1 : (float*)d_out;

        gat_gemm3<<<gGemm, 256, 0, stream>>>(xin, Wm[layer][0], Wm[layer][1], Wm[layer][2],
                                             biasv[layer], xl, xr, xres, N);
        hipMemsetAsync(agg, 0, NC * sizeof(float), stream);
        hipMemsetAsync(denom, 0, (size_t)N * sizeof(float), stream);
        gat_init_nodes<<<gNode, 256, 0, stream>>>(emax, N);

        gat_edge_logits<<<gEdgeWave, 256, 0, stream>>>(xl, xr, attv[layer], src, dst, ebuf, emax, E);
        gat_edge_exp<<<gEdgeThr, 256, 0, stream>>>(ebuf, emax, dst, denom, E);
        gat_edge_scatter<<<gEdgeWave, 256, 0, stream>>>(ebuf, denom, xl, src, dst, agg, E);
        gat_finalize<<<gElem, 256, 0, stream>>>(agg, xres, xout, (long)NC);
    }
}